// DIY_TGCN_18159121727862
// MI455X (gfx1250) — compile-verified
//
#include <hip/hip_runtime.h>
#include <math.h>

typedef float v2f __attribute__((ext_vector_type(2)));
typedef float v8f __attribute__((ext_vector_type(8)));

#define INC 128
#define HID 64

// ---------- zero deg + agg scratch ----------
__global__ void k_zero(float* __restrict__ p, long n) {
  long i = (long)blockIdx.x * blockDim.x + threadIdx.x;
  long stride = (long)gridDim.x * blockDim.x;
  for (; i < n; i += stride) p[i] = 0.0f;
}

// ---------- per-edge degree accumulation ----------
__global__ void k_deg(const long long* __restrict__ ei, const float* __restrict__ w,
                      float* __restrict__ deg, int E) {
  int e = blockIdx.x * blockDim.x + threadIdx.x;
  if (e < E) {
    int dst = (int)ei[(long)E + e];
    atomicAdd(deg + dst, w[e]);
  }
}

// ---------- dinv = rsqrt(deg + 1) (self-loop weight 1), in place ----------
__global__ void k_dinv(float* __restrict__ deg, int n) {
  int i = blockIdx.x * blockDim.x + threadIdx.x;
  if (i < n) deg[i] = rsqrtf(deg[i] + 1.0f);
}

// ---------- xw = x @ W_gcn^T via V_WMMA_F32_16X16X4_F32 ----------
// block = 128 threads (4 waves); block owns 16 rows; wave w owns output cols [16w,16w+16)
__global__ void k_gemm1(const float* __restrict__ x, const float* __restrict__ Wg,
                        float* __restrict__ xw, int n) {
  __shared__ float xs[16 * 132];            // 16 rows x 128, stride 132 (bank-safe)
  const int tid  = threadIdx.x;
  const int base = blockIdx.x * 16;
  for (int idx = tid; idx < 16 * INC; idx += 128) {
    int i = idx >> 7, k = idx & 127;
    int node = base + i;
    xs[i * 132 + k] = (node < n) ? x[(long)node * INC + k] : 0.0f;
  }
  __syncthreads();

  const int lane = tid & 31;
  const int wave = tid >> 5;                // 0..3 -> column tile
  const int col0 = wave * 16;
  const int r16  = lane & 15;               // row (A) / col (B)
  const int kh   = lane >> 4;               // K half: lanes 16-31 hold K+2,K+3
  v8f acc = {};
  for (int k0 = 0; k0 < INC; k0 += 4) {
    v2f a, b;
    a.x = xs[r16 * 132 + k0 + kh * 2];
    a.y = xs[r16 * 132 + k0 + kh * 2 + 1];
    const float* wr = Wg + (long)(col0 + r16) * INC + k0 + kh * 2;  // B[k][col]=Wg[col][k]
    b.x = wr[0];
    b.y = wr[1];
    acc = __builtin_amdgcn_wmma_f32_16x16x4_f32(false, a, false, b, (short)0, acc,
                                                false, false);
  }
  // C/D layout: VGPR r -> M = r + 8*(lane>=16), N = lane&15
  for (int r = 0; r < 8; ++r) {
    int m = base + r + kh * 8;
    if (m < n) xw[(long)m * HID + col0 + r16] = acc[r];
  }
}

// ---------- edge scatter: agg[dst] += dinv[src]*w*dinv[dst] * xw[src] ----------
// 64 consecutive threads per edge -> wave-uniform edge scalars, coalesced rows.
__global__ void k_scatter(const long long* __restrict__ ei, const float* __restrict__ w,
                          const float* __restrict__ dinv, const float* __restrict__ xw,
                          float* __restrict__ agg, int E) {
  long total  = (long)E * HID;
  long idx    = (long)blockIdx.x * blockDim.x + threadIdx.x;
  long stride = (long)gridDim.x * blockDim.x;
  for (; idx < total; idx += stride) {
    int e = (int)(idx >> 6);
    int c = (int)(idx & 63);
    int s = (int)ei[e];
    int d = (int)ei[(long)E + e];
    float coef = dinv[s] * w[e] * dinv[d];
    atomicAdd(agg + (long)d * HID + c, xw[(long)s * HID + c] * coef);
  }
}

// ---------- fused: gcn_out build + gi/gh WMMA GEMMs + GRU gates ----------
// block = 128 threads (4 waves) per 16 nodes.
__global__ void k_gru(const float* __restrict__ agg, const float* __restrict__ xw,
                      const float* __restrict__ dinv, const float* __restrict__ bg,
                      const float* __restrict__ Wih, const float* __restrict__ Whh,
                      const float* __restrict__ bih, const float* __restrict__ bhh,
                      const float* __restrict__ hmem, float* __restrict__ out, int n) {
  __shared__ float gA[16 * 66];    // gcn_out tile (16 x 64), padded
  __shared__ float hA[16 * 66];    // h_mem  tile
  __shared__ float gi[16 * 194];   // gcn_out @ W_ih^T + b_ih  (16 x 192)
  __shared__ float gh[16 * 194];   // h_mem  @ W_hh^T + b_hh
  const int tid  = threadIdx.x;
  const int base = blockIdx.x * 16;

  for (int idx = tid; idx < 16 * HID; idx += 128) {
    int i = idx >> 6, c = idx & 63;
    int node = base + i;
    float g = 0.0f, h = 0.0f;
    if (node < n) {
      float di = dinv[node];
      g = agg[(long)node * HID + c] + di * di * xw[(long)node * HID + c] + bg[c];
      h = hmem[(long)node * HID + c];
    }
    gA[i * 66 + c] = g;
    hA[i * 66 + c] = h;
  }
  __syncthreads();

  const int lane = tid & 31;
  const int wave = tid >> 5;
  const int r16  = lane & 15;
  const int kh   = lane >> 4;
  // 24 column tiles: 0..11 -> gi (A=gA, W_ih), 12..23 -> gh (A=hA, W_hh). 6 per wave.
  for (int j = 0; j < 6; ++j) {
    int tt = wave * 6 + j;                 // wave-uniform
    bool is_gh = (tt >= 12);
    int ct = is_gh ? (tt - 12) : tt;
    const float* A  = is_gh ? hA  : gA;
    const float* W  = is_gh ? Whh : Wih;
    const float* bb = is_gh ? bhh : bih;
    float* D        = is_gh ? gh  : gi;
    int col0 = ct * 16;
    v8f acc = {};
    for (int k0 = 0; k0 < HID; k0 += 4) {
      v2f a, b;
      a.x = A[r16 * 66 + k0 + kh * 2];
      a.y = A[r16 * 66 + k0 + kh * 2 + 1];
      const float* wr = W + (long)(col0 + r16) * HID + k0 + kh * 2;
      b.x = wr[0];
      b.y = wr[1];
      acc = __builtin_amdgcn_wmma_f32_16x16x4_f32(false, a, false, b, (short)0, acc,
                                                  false, false);
    }
    for (int r = 0; r < 8; ++r) {
      int m = r + kh * 8;
      D[m * 194 + col0 + r16] = acc[r] + bb[col0 + r16];
    }
  }
  __syncthreads();

  for (int idx = tid; idx < 16 * HID; idx += 128) {
    int i = idx >> 6, c = idx & 63;
    int node = base + i;
    if (node < n) {
      float rr = gi[i * 194 + c]        + gh[i * 194 + c];
      float zz = gi[i * 194 + 64 + c]   + gh[i * 194 + 64 + c];
      float nn = gi[i * 194 + 128 + c];
      float hn = gh[i * 194 + 128 + c];
      float r  = 1.0f / (1.0f + __expf(-rr));
      float z  = 1.0f / (1.0f + __expf(-zz));
      float na = tanhf(nn + r * hn);
      out[(long)node * HID + c] = (1.0f - z) * na + z * hA[i * 66 + c];
    }
  }
}

extern "C" void kernel_launch(void* const* d_in, const int* in_sizes, int n_in,
                              void* d_out, int out_size, void* d_ws, size_t ws_size,
                              hipStream_t stream) {
  const float*     x   = (const float*)d_in[0];
  const long long* ei  = (const long long*)d_in[1];   // int64 edge_index (2,E)
  const float*     w   = (const float*)d_in[2];
  const float*     Wg  = (const float*)d_in[3];
  const float*     bg  = (const float*)d_in[4];
  const float*     Wih = (const float*)d_in[5];
  const float*     Whh = (const float*)d_in[6];
  const float*     bih = (const float*)d_in[7];
  const float*     bhh = (const float*)d_in[8];
  const float*     hm  = (const float*)d_in[9];
  float* out = (float*)d_out;

  const int N = in_sizes[0] / INC;
  const int E = in_sizes[2];

  // workspace layout: xw [N*64] | deg/dinv [N] | agg [N*64]
  float* xw  = (float*)d_ws;
  float* deg = xw + (size_t)N * HID;
  float* agg = deg + N;

  const long zcount = (long)N * (HID + 1);            // deg + agg
  int zblocks = (int)((zcount + 255) / 256);
  k_zero<<<zblocks, 256, 0, stream>>>(deg, zcount);

  int tiles = (N + 15) / 16;
  k_gemm1<<<tiles, 128, 0, stream>>>(x, Wg, xw, N);

  k_deg<<<(E + 255) / 256, 256, 0, stream>>>(ei, w, deg, E);
  k_dinv<<<(N + 255) / 256, 256, 0, stream>>>(deg, N);

  k_scatter<<<8192, 256, 0, stream>>>(ei, w, deg, xw, agg, E);

  k_gru<<<tiles, 128, 0, stream>>>(agg, xw, deg, bg, Wih, Whh, bih, bhh, hm, out, N);
}